// GCNWithEdgePrediction_34918084116769
// MI455X (gfx1250) — compile-verified
//
#include <hip/hip_runtime.h>
#include <math.h>

typedef __attribute__((ext_vector_type(2))) float v2f;
typedef __attribute__((ext_vector_type(8))) float v8f;

#define TPB 256

// ---------------------------------------------------------------------------
// init: zero accumulators, amax = -inf, zero out_nodes accumulator
// ---------------------------------------------------------------------------
__global__ void k_init(float* __restrict__ s_sum, float* __restrict__ deg,
                       float* __restrict__ amax, float* __restrict__ denom,
                       float* __restrict__ out, int n) {
    long long i = (long long)blockIdx.x * blockDim.x + threadIdx.x;
    long long tot = (long long)n * 64;
    if (i < tot) out[i] = 0.0f;
    if (i < n) {
        s_sum[i] = 0.0f;
        deg[i]   = 0.0f;
        amax[i]  = -INFINITY;
        denom[i] = 0.0f;
    }
}

// ---------------------------------------------------------------------------
// w_e[k] = sum_j W_edge[k][j] * att_edge[j]   (9x64 collapse)
// ---------------------------------------------------------------------------
__global__ void k_we(const float* __restrict__ W_edge,
                     const float* __restrict__ att_edge,
                     float* __restrict__ w_e) {
    int k = threadIdx.x;
    if (k < 9) {
        float acc = 0.0f;
        #pragma unroll
        for (int j = 0; j < 64; ++j) acc += W_edge[k * 64 + j] * att_edge[j];
        w_e[k] = acc;
    }
}

// ---------------------------------------------------------------------------
// h = x @ W  via V_WMMA_F32_16X16X4_F32.  One wave per 16-row tile.
// A (16x4 f32): lanes 0-15 hold M=lane {K0,K1}; lanes 16-31 hold M=lane-16 {K2,K3}
// B (4x16 f32): VGPR0 = rows K0 (lanes 0-15) / K2 (lanes 16-31), VGPR1 = K1/K3
// C (16x16 f32): VGPR v = row (v + 8*half), col = lane&15
// Wave-uniform full-tile fast path: no per-store exec masking.
// ---------------------------------------------------------------------------
__global__ void k_gemm_h(const float* __restrict__ x, const float* __restrict__ W,
                         float* __restrict__ h, int nrows) {
    int wave = (int)(((long long)blockIdx.x * blockDim.x + threadIdx.x) >> 5);
    int lane = threadIdx.x & 31;
    int row0 = wave * 16;
    if (row0 >= nrows) return;
    const int half = lane >> 4;
    const int l16  = lane & 15;
    const bool full = (row0 + 16 <= nrows);          // wave-uniform

    int arow = row0 + l16;
    int arow_c = arow < nrows ? arow : (nrows - 1);  // clamp loads for tail tile

    // A: four 16x4 K-chunks; per-lane float2 at (row, ks*4 + 2*half)
    const v2f* xr = (const v2f*)(x + (long long)arow_c * 16 + half * 2);
    v2f a[4];
    #pragma unroll
    for (int ks = 0; ks < 4; ++ks) a[ks] = xr[ks * 2];

    #pragma unroll
    for (int j = 0; j < 4; ++j) {
        v8f c = {};
        const int col = j * 16 + l16;
        #pragma unroll
        for (int ks = 0; ks < 4; ++ks) {
            v2f b;
            const int kb = ks * 4 + half * 2;
            b.x = W[(kb + 0) * 64 + col];
            b.y = W[(kb + 1) * 64 + col];
            c = __builtin_amdgcn_wmma_f32_16x16x4_f32(
                    false, a[ks], false, b, (short)0, c, false, false);
        }
        float* hp = h + (long long)(row0 + 8 * half) * 64 + col;
        if (full) {
            #pragma unroll
            for (int v = 0; v < 8; ++v) hp[(long long)v * 64] = c[v];
        } else {
            #pragma unroll
            for (int v = 0; v < 8; ++v)
                if (row0 + 8 * half + v < nrows) hp[(long long)v * 64] = c[v];
        }
    }
}

// ---------------------------------------------------------------------------
// a_src[i] = h[i]·att_src ; a_dst[i] = h[i]·att_dst
// ---------------------------------------------------------------------------
__global__ void k_attn(const float* __restrict__ h,
                       const float* __restrict__ att_src,
                       const float* __restrict__ att_dst,
                       float* __restrict__ a_src, float* __restrict__ a_dst, int n) {
    int i = blockIdx.x * blockDim.x + threadIdx.x;
    if (i >= n) return;
    const float4* hr = (const float4*)(h + (long long)i * 64);
    const float4* as = (const float4*)att_src;
    const float4* ad = (const float4*)att_dst;
    float s1 = 0.0f, s2 = 0.0f;
    #pragma unroll
    for (int q = 0; q < 16; ++q) {
        float4 hv = hr[q], a = as[q], b = ad[q];
        s1 += hv.x * a.x + hv.y * a.y + hv.z * a.z + hv.w * a.w;
        s2 += hv.x * b.x + hv.y * b.y + hv.z * b.z + hv.w * b.w;
    }
    a_src[i] = s1;
    a_dst[i] = s2;
}

// ---------------------------------------------------------------------------
// per real edge: ae = edge_attr·w_e ; accumulate segment sum + degree at dst
// ---------------------------------------------------------------------------
__global__ void k_edge_pre(const float* __restrict__ edge_attr,
                           const int* __restrict__ dst,
                           const float* __restrict__ w_e,
                           float* __restrict__ alpha,
                           float* __restrict__ s_sum, float* __restrict__ deg, int E) {
    int e = blockIdx.x * blockDim.x + threadIdx.x;
    if (e >= E) return;
    float acc = 0.0f;
    #pragma unroll
    for (int k = 0; k < 9; ++k) acc += edge_attr[(long long)e * 9 + k] * w_e[k];
    alpha[e] = acc;
    int d = dst[e];
    atomicAdd(&s_sum[d], acc);
    atomicAdd(&deg[d], 1.0f);
}

// ---------------------------------------------------------------------------
// alpha_raw = leaky_relu(a_src[src] + a_dst[dst] + a_edge); segment max (atomic)
// flat edge ids: [0,E) real edges, [E,E+N) self-loops
// ---------------------------------------------------------------------------
__global__ void k_alpha(const int* __restrict__ src, const int* __restrict__ dst,
                        const float* __restrict__ a_src, const float* __restrict__ a_dst,
                        const float* __restrict__ s_sum, const float* __restrict__ deg,
                        float* __restrict__ alpha, float* __restrict__ amax,
                        int E, int n) {
    int f = blockIdx.x * blockDim.x + threadIdx.x;
    if (f >= E + n) return;
    int sn, dn; float ae;
    if (f < E) { sn = src[f]; dn = dst[f]; ae = alpha[f]; }
    else { int i = f - E; sn = i; dn = i; ae = s_sum[i] / fmaxf(deg[i], 1.0f); }
    float al = a_src[sn] + a_dst[dn] + ae;
    al = (al >= 0.0f) ? al : 0.2f * al;
    alpha[f] = al;
    if (al >= 0.0f) atomicMax((int*)&amax[dn], __float_as_int(al));
    else            atomicMin((unsigned int*)&amax[dn], __float_as_uint(al));
}

// ---------------------------------------------------------------------------
// ex = exp(alpha - amax[dst]); segment sum of ex (atomic)
// ---------------------------------------------------------------------------
__global__ void k_exp(const int* __restrict__ dst, const float* __restrict__ amax,
                      float* __restrict__ alpha, float* __restrict__ denom, int E, int n) {
    int f = blockIdx.x * blockDim.x + threadIdx.x;
    if (f >= E + n) return;
    int dn = (f < E) ? dst[f] : (f - E);
    float ex = expf(alpha[f] - amax[dn]);
    alpha[f] = ex;
    atomicAdd(&denom[dn], ex);
}

// ---------------------------------------------------------------------------
// normalize alpha in place
// ---------------------------------------------------------------------------
__global__ void k_norm(const int* __restrict__ dst, const float* __restrict__ denom,
                       float* __restrict__ alpha, int E, int n) {
    int f = blockIdx.x * blockDim.x + threadIdx.x;
    if (f >= E + n) return;
    int dn = (f < E) ? dst[f] : (f - E);
    alpha[f] = alpha[f] / (denom[dn] + 1e-16f);
}

// ---------------------------------------------------------------------------
// out[dst] += alpha * h[src]  — 64 threads per edge (channel = g & 63)
// ---------------------------------------------------------------------------
__global__ void k_scatter(const int* __restrict__ src, const int* __restrict__ dst,
                          const float* __restrict__ h, const float* __restrict__ alpha,
                          float* __restrict__ out, int E, int n) {
    long long g = (long long)blockIdx.x * blockDim.x + threadIdx.x;
    long long tot = (long long)(E + n) * 64;
    if (g >= tot) return;
    int f = (int)(g >> 6);
    int c = (int)(g & 63);
    int sn = (f < E) ? src[f] : (f - E);
    int dn = (f < E) ? dst[f] : (f - E);
    float v = alpha[f] * h[(long long)sn * 64 + c];
    atomicAdd(&out[(long long)dn * 64 + c], v);
}

// ---------------------------------------------------------------------------
// x2 = relu(out_nodes + bias)  (in place)
// ---------------------------------------------------------------------------
__global__ void k_relu(float* __restrict__ out, const float* __restrict__ bias,
                       long long tot) {
    long long i = (long long)blockIdx.x * blockDim.x + threadIdx.x;
    if (i >= tot) return;
    int c = (int)(i & 63);
    out[i] = fmaxf(out[i] + bias[c], 0.0f);
}

// ---------------------------------------------------------------------------
// edge readout: sigmoid( ((x2[src]+x2[dst])*0.5) · lin_w + lin_b )
// ---------------------------------------------------------------------------
__global__ void k_out(const int* __restrict__ src, const int* __restrict__ dst,
                      const float* __restrict__ x2, const float* __restrict__ lin_w,
                      const float* __restrict__ lin_b, float* __restrict__ out, int E) {
    int e = blockIdx.x * blockDim.x + threadIdx.x;
    if (e >= E) return;
    const float4* a = (const float4*)(x2 + (long long)src[e] * 64);
    const float4* b = (const float4*)(x2 + (long long)dst[e] * 64);
    const float4* w = (const float4*)lin_w;
    float acc = 0.0f;
    #pragma unroll
    for (int q = 0; q < 16; ++q) {
        float4 av = a[q], bv = b[q], wv = w[q];
        acc += (av.x + bv.x) * wv.x + (av.y + bv.y) * wv.y +
               (av.z + bv.z) * wv.z + (av.w + bv.w) * wv.w;
    }
    acc = acc * 0.5f + lin_b[0];
    out[e] = 1.0f / (1.0f + expf(-acc));
}

extern "C" void kernel_launch(void* const* d_in, const int* in_sizes, int n_in,
                              void* d_out, int out_size, void* d_ws, size_t ws_size,
                              hipStream_t stream) {
    const float* x        = (const float*)d_in[0];
    const float* edge_attr= (const float*)d_in[1];
    const float* W        = (const float*)d_in[2];
    const float* W_edge   = (const float*)d_in[3];
    const float* att_src  = (const float*)d_in[4];
    const float* att_dst  = (const float*)d_in[5];
    const float* att_edge = (const float*)d_in[6];
    const float* bias     = (const float*)d_in[7];
    const float* lin_w    = (const float*)d_in[8];
    const float* lin_b    = (const float*)d_in[9];
    const int*   ei       = (const int*)d_in[10];

    const int N = in_sizes[0] / 16;   // F_IN = 16
    const int E = in_sizes[1] / 9;    // E_DIM = 9
    const int* srcI = ei;
    const int* dstI = ei + E;

    // workspace layout (floats)
    float* ws    = (float*)d_ws;
    float* h     = ws;                        // N*64
    float* x2    = h     + (size_t)N * 64;    // N*64 (accumulator -> relu'd)
    float* a_src = x2    + (size_t)N * 64;    // N
    float* a_dst = a_src + N;                 // N
    float* s_sum = a_dst + N;                 // N
    float* deg   = s_sum + N;                 // N
    float* amax  = deg   + N;                 // N
    float* denom = amax  + N;                 // N
    float* alpha = denom + N;                 // E+N
    float* w_e   = alpha + (size_t)(E + N);   // 16

    const long long n64 = (long long)N * 64;
    const int flat = E + N;

    dim3 blk(TPB);
    int g_init    = (int)((n64 + TPB - 1) / TPB);
    int g_gemm    = (int)((((long long)((N + 15) / 16) * 32) + TPB - 1) / TPB);
    int g_node    = (N + TPB - 1) / TPB;
    int g_edge    = (E + TPB - 1) / TPB;
    int g_flat    = (flat + TPB - 1) / TPB;
    int g_scatter = (int)(((long long)flat * 64 + TPB - 1) / TPB);

    k_init<<<g_init, blk, 0, stream>>>(s_sum, deg, amax, denom, x2, N);
    k_we<<<1, 32, 0, stream>>>(W_edge, att_edge, w_e);
    k_gemm_h<<<g_gemm, blk, 0, stream>>>(x, W, h, N);
    k_attn<<<g_node, blk, 0, stream>>>(h, att_src, att_dst, a_src, a_dst, N);
    k_edge_pre<<<g_edge, blk, 0, stream>>>(edge_attr, dstI, w_e, alpha, s_sum, deg, E);
    k_alpha<<<g_flat, blk, 0, stream>>>(srcI, dstI, a_src, a_dst, s_sum, deg,
                                        alpha, amax, E, N);
    k_exp<<<g_flat, blk, 0, stream>>>(dstI, amax, alpha, denom, E, N);
    k_norm<<<g_flat, blk, 0, stream>>>(dstI, denom, alpha, E, N);
    k_scatter<<<g_scatter, blk, 0, stream>>>(srcI, dstI, h, alpha, x2, E, N);
    k_relu<<<g_init, blk, 0, stream>>>(x2, bias, n64);
    k_out<<<g_edge, blk, 0, stream>>>(srcI, dstI, x2, lin_w, lin_b, (float*)d_out, E);
}